// adjust_60327110640064
// MI455X (gfx1250) — compile-verified
//
#include <hip/hip_runtime.h>
#include <math.h>

typedef __attribute__((ext_vector_type(2))) float v2f;
typedef __attribute__((ext_vector_type(8))) float v8f;

#define HW    16384
#define WDIM  128
#define HDIM  128
#define CINCH 128
#define GRP   4
#define CG    32
#define KD    1152   // K for stages 2 and 3 (= 128*9 = G*CG*9)

__device__ __forceinline__ v8f wmma_f32(v2f a, v2f b, v8f c) {
  // V_WMMA_F32_16X16X4_F32 : D(16x16,f32) = A(16x4,f32) * B(4x16,f32) + C
  return __builtin_amdgcn_wmma_f32_16x16x4_f32(false, a, false, b, (short)0, c,
                                               false, false);
}

// ---------------------------------------------------------------------------
// Weight repacks: lay A out contiguously in the GEMM K-order so the per-WMMA
// A fragment (two consecutive K elements) is a single global_load_b64.
// ---------------------------------------------------------------------------
__global__ __launch_bounds__(256)
void repack_wom(const float* __restrict__ w, float* __restrict__ wr) {
  int idx = blockIdx.x * 256 + threadIdx.x;          // [108][1152]
  if (idx >= 108 * KD) return;
  int o = idx / KD, k = idx % KD;
  int tap = k >> 7, c = k & 127;                     // k = tap*128 + c
  wr[idx] = w[(size_t)o * KD + c * 9 + tap];         // src [o][c][tap]
}

__global__ __launch_bounds__(256)
void repack_wdcn(const float* __restrict__ w, float* __restrict__ wr) {
  int idx = blockIdx.x * 256 + threadIdx.x;          // [128][1152]
  if (idx >= CINCH * KD) return;
  int o = idx / KD, k = idx % KD;
  int gk = k >> 5, c = k & 31;                       // k = (g*9+kk)*32 + c
  int g = gk / 9, kk = gk % 9;
  wr[idx] = w[(size_t)o * KD + (g * CG + c) * 9 + kk]; // src [o][cin][kk]
}

// ---------------------------------------------------------------------------
// Kernel 1: 1x1 offset-feature conv.  off[b,o,n] = Woff[o,0:256] . X[0:256,n]
// X rows 0..127 = feat_u, rows 128..255 = 2*feat_s.
// M=128 (8 waves x 16), N=32 per block (two 16-col sub-tiles per wave).
// ---------------------------------------------------------------------------
__global__ __launch_bounds__(256)
void k1_offfeat(const float* __restrict__ fu, const float* __restrict__ fs,
                const float* __restrict__ woff, float* __restrict__ off) {
  __shared__ float sX[2 * 256 * 16];                 // 32 KB: two K=256 x 16 tiles
  const int b  = blockIdx.y;
  const int n0 = blockIdx.x * 32;
  const int tid = threadIdx.x;
  const float* fub = fu + (size_t)b * CINCH * HW;
  const float* fsb = fs + (size_t)b * CINCH * HW;

  for (int ti = tid; ti < 256 * 32; ti += 256) {
    int k = ti >> 5, col = ti & 31;
    float v = (k < CINCH) ? fub[(size_t)k * HW + n0 + col]
                          : 2.0f * fsb[(size_t)(k - CINCH) * HW + n0 + col];
    sX[(col >> 4) * (256 * 16) + k * 16 + (col & 15)] = v;
  }
  __syncthreads();

  const int wave = tid >> 5, lane = tid & 31;
  const int lo = lane & 15, half = lane >> 4;
  const int m0 = wave * 16;
  v8f acc0 = {}, acc1 = {};
  const float* wrow = woff + (size_t)(m0 + lo) * 256;
  const float* s0 = sX;
  const float* s1 = sX + 256 * 16;
  for (int k4 = 0; k4 < 256; k4 += 4) {
    int ka = k4 + 2 * half;
    v2f a, b0, b1;
    a.x  = wrow[ka];   a.y  = wrow[ka + 1];          // one b64
    b0.x = s0[ka * 16 + lo]; b0.y = s0[(ka + 1) * 16 + lo];
    b1.x = s1[ka * 16 + lo]; b1.y = s1[(ka + 1) * 16 + lo];
    acc0 = wmma_f32(a, b0, acc0);                    // A reused for both tiles
    acc1 = wmma_f32(a, b1, acc1);
  }
  float* ob = off + (size_t)b * CINCH * HW + n0 + lo;
#pragma unroll
  for (int r = 0; r < 8; ++r) {
    int o = m0 + r + 8 * half;
    ob[(size_t)o * HW]      = acc0[r];
    ob[(size_t)o * HW + 16] = acc1[r];
  }
}

// ---------------------------------------------------------------------------
// Kernel 2: 3x3 conv (pad 1) -> om[b,0:108,h,w].  Implicit im2col GEMM,
// K = tap*128 + c (repacked A), M = 108 pad 112 (waves 0..6), N=32/block.
// ---------------------------------------------------------------------------
__global__ __launch_bounds__(256)
void k2_om(const float* __restrict__ off, const float* __restrict__ womR,
           const float* __restrict__ bom, float* __restrict__ om) {
  __shared__ float sB[2 * CINCH * 16];               // 16 KB per-tap slices
  const int b  = blockIdx.y;
  const int nt = blockIdx.x;                         // 0..511
  const int y  = nt >> 2;
  const int x0 = (nt & 3) * 32;
  const int tid = threadIdx.x;
  const int wave = tid >> 5, lane = tid & 31;
  const int lo = lane & 15, half = lane >> 4;
  const int m0 = wave * 16;
  const float* offb = off + (size_t)b * CINCH * HW;
  v8f acc0 = {}, acc1 = {};

  for (int tap = 0; tap < 9; ++tap) {
    int ky = tap / 3 - 1, kx = tap % 3 - 1;
    int yy = y + ky;
    bool oky = (yy >= 0) && (yy < HDIM);
    __syncthreads();                                  // protect previous tap
    for (int ti = tid; ti < CINCH * 32; ti += 256) {
      int c = ti >> 5, col = ti & 31;
      int xx = x0 + col + kx;
      bool ok = oky && (xx >= 0) && (xx < WDIM);
      sB[(col >> 4) * (CINCH * 16) + c * 16 + (col & 15)] =
          ok ? offb[(size_t)c * HW + yy * WDIM + xx] : 0.0f;
    }
    __syncthreads();
    if (wave < 7) {                                   // wave-uniform branch
      const float* wbase = womR + (size_t)(m0 + lo) * KD + tap * CINCH;
      const float* s0 = sB;
      const float* s1 = sB + CINCH * 16;
      for (int c4 = 0; c4 < CINCH; c4 += 4) {
        int ca = c4 + 2 * half;
        v2f a, b0, b1;
        a.x  = wbase[ca]; a.y = wbase[ca + 1];        // one b64 (repacked)
        b0.x = s0[ca * 16 + lo]; b0.y = s0[(ca + 1) * 16 + lo];
        b1.x = s1[ca * 16 + lo]; b1.y = s1[(ca + 1) * 16 + lo];
        acc0 = wmma_f32(a, b0, acc0);
        acc1 = wmma_f32(a, b1, acc1);
      }
    }
  }
  if (wave < 7) {
#pragma unroll
    for (int r = 0; r < 8; ++r) {
      int o = m0 + r + 8 * half;
      if (o < 108) {
        float bo = bom[o];
        size_t p = ((size_t)b * 108 + o) * HW + y * WDIM + x0 + lo;
        om[p]      = acc0[r] + bo;
        om[p + 16] = acc1[r] + bo;
      }
    }
  }
}

// ---------------------------------------------------------------------------
// Kernel 3: deformable sampling (phase 1, cooperative -> LDS) + output GEMM
// (phase 2, WMMA, repacked A).  K = (g*9+kk)*32 + c, M = Co = 128 (8 waves).
// ---------------------------------------------------------------------------
__global__ __launch_bounds__(256)
void k3_dcn(const float* __restrict__ fs, const float* __restrict__ om,
            const float* __restrict__ wdcnR, const float* __restrict__ bdcn,
            float* __restrict__ out) {
  extern __shared__ float sS[];                      // 1152 x 16 f32 = 72 KB
  const int b  = blockIdx.y;
  const int nt = blockIdx.x;
  const int y  = nt >> 3;
  const int x0 = (nt & 7) * 16;
  const int tid = threadIdx.x;
  const float* omb = om + (size_t)b * 108 * HW + (size_t)y * WDIM;
  const float* fsb = fs + (size_t)b * CINCH * HW;

  // ---- phase 1: modulated bilinear samples for the whole K x 16 tile ----
  for (int ti = tid; ti < 36 * 16; ti += 256) {
    int col = ti & 15, gk = ti >> 4;                  // gk = g*9 + kk
    int g = gk / 9, kk = gk % 9;
    int x = x0 + col;
    float dy = omb[(size_t)gk * HW + x];
    float dx = omb[(size_t)(36 + gk) * HW + x];
    float mr = omb[(size_t)(72 + gk) * HW + x];
    float mk = 1.0f / (1.0f + __expf(-mr));           // sigmoid mask
    float py = (float)(y + kk / 3 - 1) + dy;
    float px = (float)(x + kk % 3 - 1) + dx;
    float fy = floorf(py), fx = floorf(px);
    int   iy = (int)fy,   ix = (int)fx;
    float wy = py - fy,   wx = px - fx;
    bool vy0 = (iy >= 0) && (iy < HDIM);
    bool vy1 = (iy + 1 >= 0) && (iy + 1 < HDIM);
    bool vx0 = (ix >= 0) && (ix < WDIM);
    bool vx1 = (ix + 1 >= 0) && (ix + 1 < WDIM);
    float w00 = (vy0 && vx0) ? (1.0f - wy) * (1.0f - wx) * mk : 0.0f;
    float w01 = (vy0 && vx1) ? (1.0f - wy) * wx * mk : 0.0f;
    float w10 = (vy1 && vx0) ? wy * (1.0f - wx) * mk : 0.0f;
    float w11 = (vy1 && vx1) ? wy * wx * mk : 0.0f;
    int y0c = min(max(iy, 0), HDIM - 1),  y1c = min(max(iy + 1, 0), HDIM - 1);
    int x0c = min(max(ix, 0), WDIM - 1),  x1c = min(max(ix + 1, 0), WDIM - 1);
    const float* base = fsb + (size_t)(g * CG) * HW;
    const float* p00 = base + y0c * WDIM + x0c;
    const float* p01 = base + y0c * WDIM + x1c;
    const float* p10 = base + y1c * WDIM + x0c;
    const float* p11 = base + y1c * WDIM + x1c;
    float* srow = sS + (size_t)gk * CG * 16 + col;
    for (int c = 0; c < CG; ++c) {
      size_t co = (size_t)c * HW;
      srow[c * 16] = w00 * p00[co] + w01 * p01[co] + w10 * p10[co] + w11 * p11[co];
    }
  }
  __syncthreads();

  // ---- phase 2: out[m0:m0+16, x0:x0+16] = WdcnR . sampled ----
  const int wave = tid >> 5, lane = tid & 31;
  const int lo = lane & 15, half = lane >> 4;
  const int m0 = wave * 16;
  v8f acc = {};
  const float* wrow = wdcnR + (size_t)(m0 + lo) * KD; // contiguous in K
  for (int k4 = 0; k4 < KD; k4 += 4) {
    int ka = k4 + 2 * half;
    v2f a, bv;
    a.x  = wrow[ka]; a.y = wrow[ka + 1];              // one b64 (repacked)
    bv.x = sS[ka * 16 + lo];
    bv.y = sS[(ka + 1) * 16 + lo];
    acc = wmma_f32(a, bv, acc);
  }
#pragma unroll
  for (int r = 0; r < 8; ++r) {
    int o = m0 + r + 8 * half;
    float v = acc[r] + bdcn[o];
    out[((size_t)b * CINCH + o) * HW + (size_t)y * WDIM + x0 + lo] =
        fmaxf(v, 0.0f);
  }
}

// ---------------------------------------------------------------------------
extern "C" void kernel_launch(void* const* d_in, const int* in_sizes, int n_in,
                              void* d_out, int out_size, void* d_ws, size_t ws_size,
                              hipStream_t stream) {
  const float* fu   = (const float*)d_in[0];   // feat_u  [2,128,128,128]
  const float* fs   = (const float*)d_in[1];   // feat_s  [2,128,128,128]
  const float* woff = (const float*)d_in[2];   // [128,256,1,1]
  const float* wom  = (const float*)d_in[3];   // [108,128,3,3]
  const float* bom  = (const float*)d_in[4];   // [108]
  const float* wdcn = (const float*)d_in[5];   // [128,128,3,3]
  const float* bdcn = (const float*)d_in[6];   // [128]
  float* out = (float*)d_out;

  float* off   = (float*)d_ws;                        // [2,128,HW]   16.8 MB
  float* om    = off   + (size_t)2 * CINCH * HW;      // [2,108,HW]   14.2 MB
  float* womR  = om    + (size_t)2 * 108 * HW;        // [108,1152]   0.5 MB
  float* wdcnR = womR  + (size_t)108 * KD;            // [128,1152]   0.6 MB

  repack_wom <<<dim3((108 * KD + 255) / 256), dim3(256), 0, stream>>>(wom, womR);
  repack_wdcn<<<dim3((CINCH * KD + 255) / 256), dim3(256), 0, stream>>>(wdcn, wdcnR);

  dim3 blk(256);
  k1_offfeat<<<dim3(512, 2),  blk, 0, stream>>>(fu, fs, woff, off);
  k2_om     <<<dim3(512, 2),  blk, 0, stream>>>(off, womR, bom, om);
  k3_dcn    <<<dim3(1024, 2), blk, (size_t)KD * 16 * sizeof(float), stream>>>(
      fs, om, wdcnR, bdcn, out);
}